// MyModel_47090021433402
// MI455X (gfx1250) — compile-verified
//
#include <hip/hip_runtime.h>

// ---------------------------------------------------------------------------
// PET MLEM subset update + CNN regularizer, MI455X (gfx1250, wave32, WMMA).
//
// Bandwidth analysis: w+idx dominate (67MB/subset-update, x16 = ~1.07GB HBM).
// Forward + adjoint both read w/idx; done as two passes inside ONE kernel so
// the 67MB working set stays resident in the 192MB L2 for the second pass.
// TOF matmuls (K_tof=9, padded to 16) run on V_WMMA_F32_16X16X4_F32.
// tof_w is staged into LDS with the Tensor Data Mover (tensor_load_to_lds),
// then expanded into zero-padded tables so all B-fragment reads are plain
// aligned ds_load_2addr_b64 (no exec-mask/predicated LDS loads near WMMAs).
// Voxel indices stay signed int and are consumed directly from the loads so
// the backend can match the saddr + voffset (scale_offset) gather mode.
// Scatter-add: global_atomic_add_f32 into a 256KB/image buffer (L2-resident).
// ---------------------------------------------------------------------------

#define BB 4
#define SS 4
#define LL 32768
#define KK 256
#define TT 9
#define N0 256
#define NVOX 65536
#define CH 16

typedef __attribute__((ext_vector_type(2))) float v2f;
typedef __attribute__((ext_vector_type(8))) float v8f;
typedef __attribute__((ext_vector_type(4))) unsigned int v4u;
typedef __attribute__((ext_vector_type(8))) unsigned int v8u;

// ---------------------------------------------------------------------------
// Kernel 1: fused forward-project + residual + back-project (scatter) for one
// subset s, all batch items. One wave handles a 16-LOR tile.
//
// WMMA layouts (CDNA5 ISA 7.12.2, f32 16x16x4):
//   A (16x4):  lane = M row (lane&15); VGPR v holds K = 2*(lane>>4) + v
//   B (4x16):  lane = N col (lane&15); VGPR v holds K = 2*(lane>>4) + v
//   C/D(16x16):lane = N col (lane&15); VGPR j holds M = j + 8*(lane>>4)
// ---------------------------------------------------------------------------
__global__ __launch_bounds__(256)
void pet_subset_kernel(const float* __restrict__ x,       // (B,NVOX) current image
                       const float* __restrict__ data,    // (B,S,L,T)
                       const float* __restrict__ mc,      // (B,S,L)
                       const float* __restrict__ contam,  // (B,S,L,T)
                       const float* __restrict__ norm,    // (B)
                       const float* __restrict__ w,       // (S,L,K)
                       const int*   __restrict__ idx,     // (S,L,K), values < NVOX
                       const float* __restrict__ tof,     // (K,T)
                       float* __restrict__ back,          // (B,NVOX), pre-zeroed
                       int s)
{
    __shared__ float tof_raw[KK * TT];       // TDM destination (9216 B)
    __shared__ float tof_pad[KK * 16];       // tof[k][t], t padded to 16 (16KB)
    __shared__ float tofT_pad[16 * KK];      // tof^T[t][k] padded       (16KB)
    __shared__ float r_lds[8][16][17];       // per-wave 16x16 r tile (padded)

    const int tid  = threadIdx.x;
    const int lane = tid & 31;
    const int wv   = tid >> 5;
    const int b    = blockIdx.y;

    // --- stage tof_w into LDS via the Tensor Data Mover (wave 0 issues) ----
    if (tid < 32) {
        const unsigned long long ga = (unsigned long long)(uintptr_t)tof;
        // generic LDS address: low 32 bits are the raw LDS byte offset
        const unsigned lds_off = (unsigned)(uintptr_t)&tof_raw[0];
        v4u g0;
        g0[0] = 1u;                                            // count=1, user mode
        g0[1] = lds_off;                                       // lds_addr
        g0[2] = (unsigned)(ga & 0xffffffffu);                  // global_addr[31:0]
        g0[3] = (unsigned)((ga >> 32) & 0x01ffffffu)           // global_addr[56:32]
              | (2u << 30);                                    // type=2 ("image")
        v8u g1;
        g1[0] = (2u << 16);                                    // data_size=4B, no mask
        g1[1] = ((unsigned)(KK * TT) & 0xffffu) << 16;         // tensor_dim0[15:0]
        g1[2] = ((unsigned)(KK * TT) >> 16) | (1u << 16);      // dim0 hi | tensor_dim1=1
        g1[3] = ((unsigned)(KK * TT)) << 16;                   // tile_dim0 = 2304
        g1[4] = 1u;                                            // tile_dim1=1, tile_dim2=0
        g1[5] = (unsigned)(KK * TT);                           // tensor_dim0_stride
        g1[6] = 0u;
        g1[7] = 0u;
        asm volatile("tensor_load_to_lds %0, %1" :: "s"(g0), "s"(g1) : "memory");
        __builtin_amdgcn_s_wait_tensorcnt(0);
    }
    __syncthreads();

    // expand raw tof into zero-padded direct and transposed tables
    for (int i = tid; i < KK * 16; i += 256) {
        const int k = i >> 4, t = i & 15;
        const float v = (t < TT) ? tof_raw[k * TT + t] : 0.f;
        tof_pad[i] = v;                    // [k][t16]
        tofT_pad[t * KK + k] = v;          // [t16][k]
    }
    __syncthreads();

    const int   m    = lane & 15;            // A-row / B-col / D-col index
    const int   hw   = lane >> 4;            // half-wave select
    const int   lor0 = (blockIdx.x * 8 + wv) * 16;
    const int   l    = lor0 + m;             // this lane's A-matrix LOR row

    const float nb = norm[b];
    const float* wS  = w   + (size_t)s * LL * KK;
    const int*   iS  = idx + (size_t)s * LL * KK;
    const float* xb  = x   + (size_t)b * NVOX;
    float*     backb = back + (size_t)b * NVOX;
    const size_t lsO = (size_t)(b * SS + s) * LL;
    const float* dataLS = data   + lsO * TT;
    const float* contLS = contam + lsO * TT;
    const float* mcLS   = mc     + lsO;

    const size_t rowOff = (size_t)l * KK;

    // ------------------ pass 1: forward proj(16 LOR x 16 T) -----------------
    v8f acc = {0.f, 0.f, 0.f, 0.f, 0.f, 0.f, 0.f, 0.f};
    for (int kc = 0; kc < KK; kc += 16) {
        __builtin_prefetch(wS + rowOff + kc + 128, 0, 1);   // stream-ahead
        __builtin_prefetch(iS + rowOff + kc + 128, 0, 1);
        int   i0[4], i1[4];
        float w0[4], w1[4];
#pragma unroll
        for (int u = 0; u < 4; ++u) {     // batch loads; adjacent pairs merge
            const int k0 = kc + 4 * u + 2 * hw;
            i0[u] = iS[rowOff + k0];
            i1[u] = iS[rowOff + k0 + 1];
            w0[u] = wS[rowOff + k0];
            w1[u] = wS[rowOff + k0 + 1];
        }
#pragma unroll
        for (int u = 0; u < 4; ++u) {
            const int k0 = kc + 4 * u + 2 * hw;
            const float2 bp = *(const float2*)&tofT_pad[m * KK + k0];
            v2f a, bb;
            a.x = w0[u] * (nb * xb[i0[u]]);
            a.y = w1[u] * (nb * xb[i1[u]]);
            bb.x = bp.x; bb.y = bp.y;
            acc = __builtin_amdgcn_wmma_f32_16x16x4_f32(
                      false, a, false, bb, (short)0, acc, false, false);
        }
    }

    // ------------------ residual r = mc*(1 - data/expected) -----------------
    // acc VGPR j holds proj[mr = j + 8*hw][t = m]
    const int t = m;
#pragma unroll
    for (int j = 0; j < 8; ++j) {
        const int mr = j + 8 * hw;
        const int ll = lor0 + mr;
        float rv = 0.f;
        if (t < TT) {
            const float mm = mcLS[ll];
            const float e  = mm * acc[j] + contLS[(size_t)ll * TT + t];
            rv = mm * (1.f - dataLS[(size_t)ll * TT + t] / e);
        }
        r_lds[wv][mr][t] = rv;          // transpose D-layout -> A-layout
    }
    __syncthreads();

    // backward A fragments: r(16 LOR x 16 T), K(=t)-chunks of 4
    v2f abw[4];
#pragma unroll
    for (int c = 0; c < 4; ++c) {
        const int tb = 4 * c + 2 * hw;
        abw[c].x = r_lds[wv][m][tb];
        abw[c].y = r_lds[wv][m][tb + 1];
    }

    // ------------- pass 2: gb(16L x 16k) = r @ tof^T, then scatter ----------
    for (int kc = 0; kc < KK; kc += 16) {
        const int kcol = kc + m;                     // this lane's k column
        v8f gb = {0.f, 0.f, 0.f, 0.f, 0.f, 0.f, 0.f, 0.f};
#pragma unroll
        for (int c = 0; c < 4; ++c) {
            const int tr = 4 * c + 2 * hw;           // tof^T rows (even)
            const float2 bp = *(const float2*)&tof_pad[kcol * 16 + tr];
            v2f bb; bb.x = bp.x; bb.y = bp.y;
            gb = __builtin_amdgcn_wmma_f32_16x16x4_f32(
                     false, abw[c], false, bb, (short)0, gb, false, false);
        }
        // D VGPR j holds gb[mr = j + 8*hw][k = kcol]; w/idx reload hits L2
#pragma unroll
        for (int j = 0; j < 8; ++j) {
            const int mr = j + 8 * hw;
            const size_t off = (size_t)(lor0 + mr) * KK + kcol;
            atomicAdd(&backb[iS[off]], wS[off] * gb[j]);
        }
    }
}

// ---------------------------------------------------------------------------
// Kernel 2: CNN (1->16 relu, 16->1, 3x3 SAME) fused with the MLEM combine:
//   x_new = relu(x - back/aones/norm + nw * cnn(x))
// 16x16 output tile per block, halo-2 input tile + 18x18x16 hidden in LDS.
// ---------------------------------------------------------------------------
__global__ __launch_bounds__(256)
void cnn_update_kernel(const float* __restrict__ x,      // (B,NVOX)
                       const float* __restrict__ back,   // (B,NVOX)
                       const float* __restrict__ aones,  // (B,S,NVOX)
                       const float* __restrict__ norm,   // (B)
                       const float* __restrict__ k1,     // (16,1,3,3)
                       const float* __restrict__ k2,     // (1,16,3,3)
                       const float* __restrict__ nw_p,   // scalar
                       float* __restrict__ xout,         // (B,NVOX)
                       int s)
{
    __shared__ float xt[20][20];
    __shared__ float ht[CH][18][18];
    __shared__ float k1s[CH][9], k2s[CH][9];

    const int tid = threadIdx.x;
    const int b   = blockIdx.z;
    const int ty0 = blockIdx.y * 16, tx0 = blockIdx.x * 16;
    const float* xb = x + (size_t)b * NVOX;

    if (tid < CH * 9) { k1s[tid / 9][tid % 9] = k1[tid]; k2s[tid / 9][tid % 9] = k2[tid]; }
    for (int i = tid; i < 20 * 20; i += 256) {
        const int yy = ty0 - 2 + i / 20, xx = tx0 - 2 + i % 20;
        xt[i / 20][i % 20] =
            (yy >= 0 && yy < N0 && xx >= 0 && xx < N0) ? xb[yy * N0 + xx] : 0.f;
    }
    __syncthreads();

    // hidden layer h on 18x18 (halo 1), zero outside image (conv2 zero-pads h)
    for (int i = tid; i < 18 * 18 * CH; i += 256) {
        const int ch = i / 324, rem = i % 324, hy = rem / 18, hx = rem % 18;
        const int gy = ty0 - 1 + hy, gx = tx0 - 1 + hx;
        float a = 0.f;
        if (gy >= 0 && gy < N0 && gx >= 0 && gx < N0) {
#pragma unroll
            for (int dy = 0; dy < 3; ++dy)
#pragma unroll
                for (int dx = 0; dx < 3; ++dx)
                    a += k1s[ch][dy * 3 + dx] * xt[hy + dy][hx + dx];
            a = fmaxf(a, 0.f);
        }
        ht[ch][hy][hx] = a;
    }
    __syncthreads();

    const int py = tid / 16, px = tid % 16;
    float x2 = 0.f;
#pragma unroll
    for (int ch = 0; ch < CH; ++ch)
#pragma unroll
        for (int dy = 0; dy < 3; ++dy)
#pragma unroll
            for (int dx = 0; dx < 3; ++dx)
                x2 += k2s[ch][dy * 3 + dx] * ht[ch][py + dy][px + dx];

    const size_t pix = (size_t)(ty0 + py) * N0 + (tx0 + px);
    const float nb  = norm[b];
    const float ao  = aones[(size_t)(b * SS + s) * NVOX + pix];
    const float x1  = back[(size_t)b * NVOX + pix] / ao / nb;
    const float v   = xb[pix] - x1 + nw_p[0] * x2;
    xout[(size_t)b * NVOX + pix] = fmaxf(v, 0.f);
}

__global__ void zero_kernel(float* __restrict__ p, int n)
{
    const int i = blockIdx.x * blockDim.x + threadIdx.x;
    if (i < n) p[i] = 0.f;
}

// ---------------------------------------------------------------------------
extern "C" void kernel_launch(void* const* d_in, const int* in_sizes, int n_in,
                              void* d_out, int out_size, void* d_ws, size_t ws_size,
                              hipStream_t stream)
{
    const float* osem   = (const float*)d_in[0];
    const float* data   = (const float*)d_in[1];
    const float* mc     = (const float*)d_in[2];
    const float* contam = (const float*)d_in[3];
    const float* aones  = (const float*)d_in[4];
    const float* norm   = (const float*)d_in[5];
    const float* w      = (const float*)d_in[6];
    const int*   idx    = (const int*)  d_in[7];
    const float* tof    = (const float*)d_in[8];
    const float* k1     = (const float*)d_in[9];
    const float* k2     = (const float*)d_in[10];
    const float* nw     = (const float*)d_in[11];
    float* out = (float*)d_out;

    float* back = (float*)d_ws;              // B*NVOX
    float* xA   = back + BB * NVOX;          // B*NVOX
    float* xB   = xA   + BB * NVOX;          // B*NVOX  (3 MB total)

    const float* xin = osem;
    for (int s = 0; s < SS; ++s) {
        float* xout = (s == SS - 1) ? out : ((s & 1) ? xB : xA);

        zero_kernel<<<(BB * NVOX + 255) / 256, 256, 0, stream>>>(back, BB * NVOX);

        pet_subset_kernel<<<dim3(LL / 128, BB), 256, 0, stream>>>(
            xin, data, mc, contam, norm, w, idx, tof, back, s);

        cnn_update_kernel<<<dim3(N0 / 16, N0 / 16, BB), 256, 0, stream>>>(
            xin, back, aones, norm, k1, k2, nw, xout, s);

        xin = xout;
    }
}